// GATv2_86904368267289
// MI455X (gfx1250) — compile-verified
//
#include <hip/hip_runtime.h>
#include <hip/hip_bf16.h>
#include <math.h>

// ---------------------------------------------------------------------------
// MI455X (gfx1250) GATv2 forward.
// Roofline: dominated by streaming Wn1/Wn2 (2 x 400MB fp32) once ~= 34us at
// 23.3 TB/s. Big GEMM is M=8 (pad 16) x N=10000 x K=10000 -> memory bound,
// keep fp32, use V_WMMA_F32_16X16X4_F32. inv chunk staged to LDS via TDM,
// shared by 5 waves per block (625 tiles = 125 blocks x 5 waves).
// ---------------------------------------------------------------------------

typedef float v2f __attribute__((ext_vector_type(2)));
typedef float v8f __attribute__((ext_vector_type(8)));
typedef unsigned int u32x4 __attribute__((ext_vector_type(4)));
typedef int i32x4 __attribute__((ext_vector_type(4)));
typedef int i32x8 __attribute__((ext_vector_type(8)));

#if defined(__has_builtin)
#if __has_builtin(__builtin_amdgcn_wmma_f32_16x16x4_f32)
#define HAS_WMMA_F32X4 1
#endif
#if __has_builtin(__builtin_amdgcn_tensor_load_to_lds) && __has_builtin(__builtin_amdgcn_s_wait_tensorcnt)
#define HAS_TDM 1
#endif
#endif

#define NB 8          // batch
#define NN 100        // nodes
#define N2 10000      // NN*NN
#define DH 128        // hidden
#define KCHUNK 1000   // K split chunk for big GEMM
#define NSPLIT 10     // N2 / KCHUNK
#define TPB 5         // r-tiles (waves) per big-GEMM block

// WMMA fp32 16x16x4 fragment mapping (ISA 7.12.2):
//  A (16x4):  lane L (L<16) holds row M=L: VGPR0=K0, VGPR1=K1; lane L+16: K2,K3
//             -> per-lane contiguous b64 load at (row L%16, k0 + (L>>4)*2)
//  B (4x16):  symmetric per-lane b64 at (col L%16, k0 + (L>>4)*2) of W[N,K]
//  D (16x16): lane L, VGPR v -> row M = v + (L>>4)*8, col N = L%16
__device__ __forceinline__ v8f wmma_f32_step(v2f a, v2f b, v8f c) {
#if HAS_WMMA_F32X4
  return __builtin_amdgcn_wmma_f32_16x16x4_f32(false, a, false, b, (short)0, c,
                                               false, false);
#else
  c[0] += a[0] * b[0] + a[1] * b[1];  // compile-only fallback
  return c;
#endif
}

// ---------------------------------------------------------------------------
// Generic small GEMM: out[M,N] = act(A[M,K] @ W[N,K]^T + bias)
// act: 0=none 1=tanh 2=relu.  One 16x16 tile per wave, 8 waves/block.
// ---------------------------------------------------------------------------
__global__ __launch_bounds__(256) void k_wmma_linear(
    const float* __restrict__ A, const float* __restrict__ W,
    const float* __restrict__ bias, float* __restrict__ out,
    int M, int N, int K, int act) {
  int Mt = (M + 15) >> 4, Nt = (N + 15) >> 4;
  int tile = blockIdx.x * 8 + (threadIdx.x >> 5);
  if (tile >= Mt * Nt) return;  // wave-uniform exit (EXEC all-ones for WMMA)
  int mt = tile / Nt, nt = tile - mt * Nt;
  int m0 = mt << 4, n0 = nt << 4;
  int lane = threadIdx.x & 31;
  int r = lane & 15, kh = (lane >> 4) << 1;
  int mrow = m0 + r; if (mrow > M - 1) mrow = M - 1;   // clamp (reads valid mem)
  int nrow = n0 + r; if (nrow > N - 1) nrow = N - 1;
  const float* ap = A + (size_t)mrow * K + kh;
  const float* wp = W + (size_t)nrow * K + kh;
  v8f acc = {};
  for (int k0 = 0; k0 < K; k0 += 4) {
    v2f a = *(const v2f*)(ap + k0);
    v2f b = *(const v2f*)(wp + k0);
    acc = wmma_f32_step(a, b, acc);
  }
  int n = n0 + (lane & 15);
  int mb = m0 + ((lane >> 4) << 3);
  for (int v = 0; v < 8; ++v) {
    int m = mb + v;
    if (m < M && n < N) {
      float x = acc[v];
      if (bias) x += bias[n];
      if (act == 1) x = tanhf(x);
      else if (act == 2) x = fmaxf(x, 0.f);
      out[(size_t)m * N + n] = x;
    }
  }
}

// ---------------------------------------------------------------------------
// Big GEMM: partial[split][rc][b16] = sum_{k in split} Wn[rc,k] * inv[b,k]
// A = Wn rows (M = rc), B = inv (batches, padded 8->16 by clamping).
// inv K-chunk staged once per block via the Tensor Data Mover, consumed by
// 5 waves (one 16-row tile each).  Grid (125, NSPLIT), block 160.
// ---------------------------------------------------------------------------
__global__ __launch_bounds__(160) void k_big_gemm(
    const float* __restrict__ Wn, const float* __restrict__ invp,
    float* __restrict__ partial) {
  __shared__ float lds_inv[NB * KCHUNK];  // 32 KB
  const int wave = threadIdx.x >> 5;
  const int lane = threadIdx.x & 31;
  const int r0 = (blockIdx.x * TPB + wave) << 4;
  const int split = blockIdx.y;
  const int ks = split * KCHUNK;

  if (wave == 0) {
#if HAS_TDM
    // Tensor DMA descriptor (CDNA5 ISA): 2D tile,
    // dim0 = K (contiguous, data_size=4B), dim1 = batch, stride0 = N2.
    unsigned long long ga = (unsigned long long)(const void*)(invp + ks);
    unsigned lds_off = (unsigned)(unsigned long long)(void*)&lds_inv[0];
    u32x4 g0 = {1u /*count=1*/, lds_off, (unsigned)(ga & 0xffffffffu),
                (unsigned)((ga >> 32) & 0x1ffffffu) | (2u << 30) /*type=2*/};
    unsigned long long tdim0 = (unsigned long long)(N2 - ks);
    unsigned tdim1 = NB, tile0 = KCHUNK, tile1 = NB, tile2 = 0;
    unsigned long long st0 = (unsigned long long)N2, st1 = 0;
    i32x8 g1 = {
        (int)(2u << 16),                                          // data_size=4B
        (int)((tdim0 & 0xffffu) << 16),                           // tdim0 lo
        (int)(((tdim0 >> 16) & 0xffffu) | ((tdim1 & 0xffffu) << 16)),
        (int)(((tdim1 >> 16) & 0xffffu) | ((tile0 & 0xffffu) << 16)),
        (int)((tile1 & 0xffffu) | ((tile2 & 0xffffu) << 16)),
        (int)(st0 & 0xffffffffu),
        (int)(((st0 >> 32) & 0xffffu) | ((st1 & 0xffffu) << 16)),
        (int)((st1 >> 16) & 0xffffffffu)};
    i32x4 zz = {0, 0, 0, 0};
    i32x8 zz8 = {0, 0, 0, 0, 0, 0, 0, 0};
    __builtin_amdgcn_tensor_load_to_lds(g0, g1, zz, zz, zz8, 0);
    __builtin_amdgcn_s_wait_tensorcnt(0);
#else
    for (int idx = lane; idx < NB * KCHUNK; idx += 32)
      lds_inv[idx] = invp[(idx / KCHUNK) * N2 + ks + (idx % KCHUNK)];
#endif
  }
  __syncthreads();

  const int r = lane & 15, kh = (lane >> 4) << 1;
  const float* arow = Wn + (size_t)(r0 + r) * N2 + ks + kh;       // global b64/step
  const float* brow = lds_inv + ((r < NB) ? r : (NB - 1)) * KCHUNK + kh;  // ds b64
  v8f acc = {};
  for (int k0 = 0; k0 < KCHUNK; k0 += 4) {
    if ((k0 & 127) == 0) __builtin_prefetch(arow + k0 + 256, 0, 1);
    v2f a = *(const v2f*)(arow + k0);
    v2f b = *(const v2f*)(brow + k0);
    acc = wmma_f32_step(a, b, acc);
  }
  // D: row m = v + (lane>>4)*8 (local rc), col n = lane&15 (batch)
  const int n = lane & 15;
  const int mb = (lane >> 4) << 3;
  float* prow = partial + ((size_t)split * N2 + r0) * 16;
  for (int v = 0; v < 8; ++v) prow[(size_t)(mb + v) * 16 + n] = acc[v];
}

// Reduce split-K partials; edge_att = tanh(sum)*mask; ef = e*edge_att; 0 -> -1e4
__global__ void k_edge_combine(const float* __restrict__ partial,
                               const float* __restrict__ mask,
                               const float* __restrict__ e,
                               float* __restrict__ ef) {
  int idx = blockIdx.x * blockDim.x + threadIdx.x;  // b*N2 + rc
  if (idx >= NB * N2) return;
  int b = idx / N2, rc = idx - b * N2;
  float s = 0.f;
  for (int sp = 0; sp < NSPLIT; ++sp)
    s += partial[((size_t)sp * N2 + rc) * 16 + b];
  float eatt = tanhf(s) * mask[idx];
  float v = e[idx] * eatt;
  ef[idx] = (v == 0.f) ? -10000.f : v;
}

// e[b,i,j] = sum_d tanh(g[b,i,d] + g[b,j,d]) * Wa[d]   (one wave per (b,i,j))
__global__ __launch_bounds__(256) void k_attn_e(const float* __restrict__ g,
                                                const float* __restrict__ Wa,
                                                float* __restrict__ e) {
  int wid = (blockIdx.x * blockDim.x + threadIdx.x) >> 5;
  int lane = threadIdx.x & 31;
  if (wid >= NB * N2) return;
  int b = wid / N2, rem = wid - b * N2;
  int i = rem / NN, j = rem - i * NN;
  const float* gi = g + (size_t)(b * NN + i) * DH;
  const float* gj = g + (size_t)(b * NN + j) * DH;
  float s = 0.f;
  for (int d = lane; d < DH; d += 32) s += tanhf(gi[d] + gj[d]) * Wa[d];
  for (int o = 16; o; o >>= 1) s += __shfl_xor(s, o, 32);
  if (lane == 0) e[wid] = s;
}

// adjacency normalization -> mask, 1/adj_n (one wave per (b,i) row)
__global__ __launch_bounds__(256) void k_adj_prep(const float* __restrict__ adj,
                                                  float* __restrict__ invp,
                                                  float* __restrict__ maskp) {
  int wid = (blockIdx.x * blockDim.x + threadIdx.x) >> 5;
  int lane = threadIdx.x & 31;
  if (wid >= NB * NN) return;
  int i = wid % NN;
  const float* row = adj + (size_t)wid * NN;
  float mn = 3.4e38f;
  for (int j = lane; j < NN; j += 32) {
    float v = row[j];
    if (v != 0.f) mn = fminf(mn, v);
  }
  for (int o = 16; o; o >>= 1) mn = fminf(mn, __shfl_xor(mn, o, 32));
  float dmin = mn * 0.5f;
  float ss = 0.f;
  for (int j = lane; j < NN; j += 32) {
    float v = row[j] + ((j == i) ? dmin : 0.f);
    ss += v * v;
  }
  for (int o = 16; o; o >>= 1) ss += __shfl_xor(ss, o, 32);
  float nrm = sqrtf(ss);
  if (nrm < 1e-12f) nrm = 1e-12f;
  float rn = 1.f / nrm;
  for (int j = lane; j < NN; j += 32) {
    float v = (row[j] + ((j == i) ? dmin : 0.f)) * rn;
    size_t o2 = (size_t)wid * NN + j;
    maskp[o2] = (v != 0.f) ? 1.f : 0.f;
    invp[o2] = (v != 0.f) ? 1.f / v : 0.f;
  }
}

// softmax over j (100) then aggregate: out[b,i,:] = tanh(sum_j a_j g[b,j,:])
__global__ __launch_bounds__(256) void k_softmax_agg(const float* __restrict__ ef,
                                                     const float* __restrict__ g,
                                                     float* __restrict__ out) {
  int wid = (blockIdx.x * blockDim.x + threadIdx.x) >> 5;
  int lane = threadIdx.x & 31;
  if (wid >= NB * NN) return;
  int b = wid / NN;
  const float* row = ef + (size_t)wid * NN;
  float mx = -3.4e38f;
  for (int j = lane; j < NN; j += 32) mx = fmaxf(mx, row[j]);
  for (int o = 16; o; o >>= 1) mx = fmaxf(mx, __shfl_xor(mx, o, 32));
  float sm = 0.f;
  for (int j = lane; j < NN; j += 32) sm += expf(row[j] - mx);
  for (int o = 16; o; o >>= 1) sm += __shfl_xor(sm, o, 32);
  float rsm = 1.f / sm;
  float a0 = 0.f, a1 = 0.f, a2 = 0.f, a3 = 0.f;
  for (int j = 0; j < NN; ++j) {
    float a = expf(row[j] - mx) * rsm;
    const float* gr = g + (size_t)(b * NN + j) * DH;
    a0 += a * gr[lane];
    a1 += a * gr[lane + 32];
    a2 += a * gr[lane + 64];
    a3 += a * gr[lane + 96];
  }
  float* orow = out + (size_t)wid * DH;
  orow[lane] = tanhf(a0);
  orow[lane + 32] = tanhf(a1);
  orow[lane + 64] = tanhf(a2);
  orow[lane + 96] = tanhf(a3);
}

__global__ void k_concat(const float* __restrict__ a, const float* __restrict__ b,
                         float* __restrict__ out, int M, int Da, int Db) {
  int idx = blockIdx.x * blockDim.x + threadIdx.x;
  int D = Da + Db;
  if (idx >= M * D) return;
  int m = idx / D, c = idx - m * D;
  out[idx] = (c < Da) ? a[(size_t)m * Da + c] : b[(size_t)m * Db + (c - Da)];
}

__global__ void k_final_softmax(const float* __restrict__ q, float* __restrict__ out,
                                int rows) {
  int r = blockIdx.x * blockDim.x + threadIdx.x;
  if (r >= rows) return;
  float a = q[2 * r], b = q[2 * r + 1];
  float m = fmaxf(a, b);
  float ea = expf(a - m), eb = expf(b - m);
  float s = ea + eb;
  out[2 * r] = ea / s;
  out[2 * r + 1] = eb / s;
}

// ---------------------------------------------------------------------------
static inline void lin(hipStream_t s, const float* A, const float* W,
                       const float* b, float* out, int M, int N, int K, int act) {
  int tiles = ((M + 15) / 16) * ((N + 15) / 16);
  int blocks = (tiles + 7) / 8;
  k_wmma_linear<<<blocks, 256, 0, s>>>(A, W, b, out, M, N, K, act);
}

extern "C" void kernel_launch(void* const* d_in, const int* in_sizes, int n_in,
                              void* d_out, int out_size, void* d_ws, size_t ws_size,
                              hipStream_t stream) {
  (void)in_sizes; (void)n_in; (void)out_size; (void)ws_size;
  const float* x    = (const float*)d_in[0];
  const float* adj  = (const float*)d_in[1];
  const float* W_in = (const float*)d_in[2];
  const float* b_in = (const float*)d_in[3];
  const float* Wl1  = (const float*)d_in[4];
  const float* Wa1  = (const float*)d_in[5];
  const float* Wn1  = (const float*)d_in[6];
  const float* W2   = (const float*)d_in[7];
  const float* b2   = (const float*)d_in[8];
  const float* Wl2  = (const float*)d_in[9];
  const float* Wa2  = (const float*)d_in[10];
  const float* Wn2  = (const float*)d_in[11];
  const float* Wm1  = (const float*)d_in[12];
  const float* bm1  = (const float*)d_in[13];
  const float* Wm2  = (const float*)d_in[14];
  const float* bm2  = (const float*)d_in[15];
  const float* Wm3  = (const float*)d_in[16];
  const float* bm3  = (const float*)d_in[17];
  float* outp = (float*)d_out;

  const int M = NB * NN;  // 800 rows
  float* ws = (float*)d_ws;
  float* h_in = ws;               ws += (size_t)M * DH;       // 800x128
  float* g1   = ws;               ws += (size_t)M * DH;
  float* e    = ws;               ws += (size_t)NB * N2;
  float* invb = ws;               ws += (size_t)NB * N2;
  float* maskb= ws;               ws += (size_t)NB * N2;
  float* part = ws;               ws += (size_t)NSPLIT * N2 * 16;  // 6.4MB
  float* ef   = ws;               ws += (size_t)NB * N2;
  float* out1 = ws;               ws += (size_t)M * DH;
  float* cat1 = ws;               ws += (size_t)M * 2 * DH;
  float* o1   = ws;               ws += (size_t)M * 2 * DH;
  float* g2   = ws;               ws += (size_t)M * DH;
  float* out2 = ws;               ws += (size_t)M * DH;
  float* cat2 = ws;               ws += (size_t)M * 3 * DH;
  float* q1   = ws;               ws += (size_t)M * 2 * DH;
  float* q2   = ws;               ws += (size_t)M * DH;
  float* q3   = ws;               ws += (size_t)M * 2;

  dim3 bigGrid(N2 / (16 * TPB), NSPLIT);  // (125, 10), block 160 = 5 waves

  // h_in = x @ W_in^T + b_in
  lin(stream, x, W_in, b_in, h_in, M, DH, 64, 0);
  // adjacency prep (same for both layers)
  k_adj_prep<<<(M * 32 + 255) / 256, 256, 0, stream>>>(adj, invb, maskb);

  // ---- GAT layer 1 ----
  lin(stream, h_in, Wl1, nullptr, g1, M, DH, DH, 0);
  k_attn_e<<<(NB * N2 * 32 + 255) / 256, 256, 0, stream>>>(g1, Wa1, e);
  k_big_gemm<<<bigGrid, 160, 0, stream>>>(Wn1, invb, part);
  k_edge_combine<<<(NB * N2 + 255) / 256, 256, 0, stream>>>(part, maskb, e, ef);
  k_softmax_agg<<<(M * 32 + 255) / 256, 256, 0, stream>>>(ef, g1, out1);
  k_concat<<<(M * 2 * DH + 255) / 256, 256, 0, stream>>>(out1, h_in, cat1, M, DH, DH);
  lin(stream, cat1, W2, b2, o1, M, 2 * DH, 2 * DH, 1);

  // ---- GAT layer 2 ----
  lin(stream, o1, Wl2, nullptr, g2, M, DH, 2 * DH, 0);
  k_attn_e<<<(NB * N2 * 32 + 255) / 256, 256, 0, stream>>>(g2, Wa2, e);
  k_big_gemm<<<bigGrid, 160, 0, stream>>>(Wn2, invb, part);
  k_edge_combine<<<(NB * N2 + 255) / 256, 256, 0, stream>>>(part, maskb, e, ef);
  k_softmax_agg<<<(M * 32 + 255) / 256, 256, 0, stream>>>(ef, g2, out2);
  k_concat<<<(M * 3 * DH + 255) / 256, 256, 0, stream>>>(out2, o1, cat2, M, DH, 2 * DH);

  // ---- MLP head ----
  lin(stream, cat2, Wm1, bm1, q1, M, 2 * DH, 3 * DH, 2);
  lin(stream, q1, Wm2, bm2, q2, M, DH, 2 * DH, 2);
  lin(stream, q2, Wm3, bm3, q3, M, 2, DH, 0);
  k_final_softmax<<<(M + 255) / 256, 256, 0, stream>>>(q3, outp, M);
}